// WideAndDeep_3212635538193
// MI455X (gfx1250) — compile-verified
//
#include <hip/hip_runtime.h>
#include <math.h>

typedef __attribute__((ext_vector_type(2))) float v2f;
typedef __attribute__((ext_vector_type(8))) float v8f;

#define BATCH 16384
#define EMB 64
#define HID1 256
#define HID2 128
#define BN_EPS 1e-5f

// ---------------- zero the global stats accumulators ----------------
__global__ void k_init(float* stats, int n) {
    int i = blockIdx.x * blockDim.x + threadIdx.x;
    if (i < n) stats[i] = 0.f;
}

// ---------------- embedding gather + wide path ----------------
// 8 rows per block of 256; 32 lanes per row; one float4 per lane (128 floats/row)
__global__ __launch_bounds__(256) void k_gather(
    const int* __restrict__ ui, const int* __restrict__ ii,
    const float* __restrict__ ut, const float* __restrict__ it,
    const float* __restrict__ ww, float* __restrict__ x,
    float* __restrict__ wide, int num_users) {
    int t = threadIdx.x;
    int r = t >> 5, lane = t & 31;
    int m = blockIdx.x * 8 + r;
    int u = ui[m], v = ii[m];
    float4 d;
    if (lane < 16) d = ((const float4*)(ut + (size_t)u * EMB))[lane];
    else           d = ((const float4*)(it + (size_t)v * EMB))[lane - 16];
    ((float4*)(x + (size_t)m * 128))[lane] = d;
    if (lane == 0) wide[m] = ww[u] + ww[num_users + v];
}

// ---------------- WMMA GEMM: H = relu(A(BATCH,K) @ W(N,K)^T + bias) ----------------
// Also accumulates per-column sum / sumsq (batch-norm statistics).
template <int N, int K>
__global__ __launch_bounds__(256) void k_gemm(
    const float* __restrict__ A, const float* __restrict__ W,
    const float* __restrict__ bias, float* __restrict__ Hout,
    float* __restrict__ gsum, float* __restrict__ gsq) {
    constexpr int BM = 64;          // rows per block
    constexpr int BK = 128;         // K chunk staged in LDS
    constexpr int KP = BK + 2;      // pad: row stride 130 -> conflict-free b64 reads
    constexpr int TPW = N / 32;     // 16x16 tiles per wave (4 mtiles * N/16 ntiles / 8 waves)

    __shared__ float sA[BM * KP];
    __shared__ float ssum[N], ssq[N];

    int t = threadIdx.x;
    int w = t >> 5, l = t & 31;
    int lh = l >> 4, lc = l & 15;
    int m0 = blockIdx.x * BM;

    for (int i = t; i < N; i += 256) { ssum[i] = 0.f; ssq[i] = 0.f; }

    v8f acc[TPW] = {};

    for (int kc = 0; kc < K; kc += BK) {
        __syncthreads();
        // cooperative stage of A chunk (BM x BK) into padded LDS, float2 granularity
        for (int i = t; i < BM * (BK / 2); i += 256) {
            int row = i / (BK / 2);
            int c2  = i % (BK / 2);
            float2 d = *(const float2*)(A + (size_t)(m0 + row) * K + kc + 2 * c2);
            *(float2*)(sA + row * KP + 2 * c2) = d;
        }
        __syncthreads();

#pragma unroll
        for (int i = 0; i < TPW; i++) {
            int tid = w * TPW + i;
            int mt = tid & 3, nt = tid >> 2;
            const float* arow = sA + (mt * 16 + lc) * KP + 2 * lh;
            const float* brow = W + (size_t)(nt * 16 + lc) * K + kc + 2 * lh;
            v8f c = acc[i];
#pragma unroll
            for (int k0 = 0; k0 < BK; k0 += 4) {
                v2f a = *(const v2f*)(arow + k0);
                v2f b = *(const v2f*)(brow + k0);
                c = __builtin_amdgcn_wmma_f32_16x16x4_f32(
                        false, a, false, b, (short)0, c, false, false);
            }
            acc[i] = c;
        }
    }

    // epilogue: bias + relu, store, local BN-stat reduction
#pragma unroll
    for (int i = 0; i < TPW; i++) {
        int tid = w * TPW + i;
        int mt = tid & 3, nt = tid >> 2;
        int col = nt * 16 + lc;
        float bn = bias[col];
        float s = 0.f, q = 0.f;
        int rowbase = m0 + mt * 16 + 8 * lh;
#pragma unroll
        for (int v = 0; v < 8; v++) {
            float h = acc[i][v] + bn;
            h = h > 0.f ? h : 0.f;
            Hout[(size_t)(rowbase + v) * N + col] = h;
            s += h;
            q += h * h;
        }
        atomicAdd(&ssum[col], s);   // ds_add_f32
        atomicAdd(&ssq[col], q);
    }
    __syncthreads();
    for (int i = t; i < N; i += 256) {
        atomicAdd(&gsum[i], ssum[i]);   // one global atomic per column per block
        atomicAdd(&gsq[i],  ssq[i]);
    }
}

// ---------------- finalize BN1 and fold its affine into W2/b2 ----------------
__global__ void k_mid1(const float* __restrict__ sum, const float* __restrict__ sq,
                       const float* __restrict__ g, const float* __restrict__ be,
                       const float* __restrict__ W2, const float* __restrict__ b2,
                       float* __restrict__ W2p, float* __restrict__ b2p) {
    __shared__ float a1[HID1], c1[HID1];
    int t = threadIdx.x;  // 256 threads
    float mean = sum[t] * (1.f / BATCH);
    float var  = sq[t] * (1.f / BATCH) - mean * mean;
    float a    = g[t] * rsqrtf(var + BN_EPS);
    a1[t] = a;
    c1[t] = be[t] - a * mean;
    __syncthreads();
    for (int n = 0; n < HID2; n++)
        W2p[n * HID1 + t] = a1[t] * W2[n * HID1 + t];
    if (t < HID2) {
        float acc = b2[t];
        for (int k = 0; k < HID1; k++) acc += c1[k] * W2[t * HID1 + k];
        b2p[t] = acc;
    }
}

// ---------------- finalize BN2 and fold into W3/b3 (+ wide bias) ----------------
__global__ void k_mid2(const float* __restrict__ sum, const float* __restrict__ sq,
                       const float* __restrict__ g, const float* __restrict__ be,
                       const float* __restrict__ W3, const float* __restrict__ b3,
                       const float* __restrict__ wide_b,
                       float* __restrict__ W3p, float* __restrict__ b3pp) {
    __shared__ float c2s[HID2];
    int t = threadIdx.x;  // 128 threads
    float mean = sum[t] * (1.f / BATCH);
    float var  = sq[t] * (1.f / BATCH) - mean * mean;
    float a    = g[t] * rsqrtf(var + BN_EPS);
    c2s[t] = be[t] - a * mean;
    W3p[t] = a * W3[t];
    __syncthreads();
    if (t == 0) {
        float acc = b3[0] + wide_b[0];
        for (int n = 0; n < HID2; n++) acc += c2s[n] * W3[n];
        b3pp[0] = acc;
    }
}

// ---------------- final: out = sigmoid(wide + h2 . W3' + b') ----------------
__global__ __launch_bounds__(256) void k_final(
    const float* __restrict__ h2, const float* __restrict__ W3p,
    const float* __restrict__ b3pp, const float* __restrict__ wide,
    float* __restrict__ out) {
    __shared__ float w3s[HID2];
    int t = threadIdx.x;
    if (t < HID2) w3s[t] = W3p[t];
    __syncthreads();
    int m = blockIdx.x * 256 + t;
    const float4* row = (const float4*)(h2 + (size_t)m * HID2);
    float acc = 0.f;
#pragma unroll
    for (int i = 0; i < HID2 / 4; i++) {
        float4 h = row[i];
        acc += h.x * w3s[4 * i] + h.y * w3s[4 * i + 1] +
               h.z * w3s[4 * i + 2] + h.w * w3s[4 * i + 3];
    }
    float z = acc + wide[m] + b3pp[0];
    out[m] = 1.f / (1.f + expf(-z));
}

extern "C" void kernel_launch(void* const* d_in, const int* in_sizes, int n_in,
                              void* d_out, int out_size, void* d_ws, size_t ws_size,
                              hipStream_t stream) {
    const int*   ui  = (const int*)d_in[0];
    const int*   ii  = (const int*)d_in[1];
    const float* ww  = (const float*)d_in[2];
    const float* wb  = (const float*)d_in[3];
    const float* ut  = (const float*)d_in[4];
    const float* it  = (const float*)d_in[5];
    const float* W1  = (const float*)d_in[6];
    const float* b1  = (const float*)d_in[7];
    const float* g1  = (const float*)d_in[8];
    const float* be1 = (const float*)d_in[9];
    const float* W2  = (const float*)d_in[10];
    const float* b2  = (const float*)d_in[11];
    const float* g2  = (const float*)d_in[12];
    const float* be2 = (const float*)d_in[13];
    const float* W3  = (const float*)d_in[14];
    const float* b3  = (const float*)d_in[15];
    int num_users = in_sizes[4] / EMB;

    float* ws   = (float*)d_ws;
    float* x    = ws;                          // 16384*128
    float* h1   = x    + (size_t)BATCH * 128;  // 16384*256
    float* h2   = h1   + (size_t)BATCH * 256;  // 16384*128
    float* wide = h2   + (size_t)BATCH * 128;  // 16384
    float* sum1 = wide + BATCH;                // 256
    float* sq1  = sum1 + 256;                  // 256
    float* sum2 = sq1  + 256;                  // 128
    float* sq2  = sum2 + 128;                  // 128
    float* W2p  = sq2  + 128;                  // 128*256
    float* b2p  = W2p  + HID2 * HID1;          // 128
    float* W3p  = b2p  + HID2;                 // 128
    float* b3pp = W3p  + HID2;                 // 1

    k_init<<<3, 256, 0, stream>>>(sum1, 768);  // sum1..sq2 contiguous
    k_gather<<<BATCH / 8, 256, 0, stream>>>(ui, ii, ut, it, ww, x, wide, num_users);
    k_gemm<HID1, 128><<<BATCH / 64, 256, 0, stream>>>(x, W1, b1, h1, sum1, sq1);
    k_mid1<<<1, 256, 0, stream>>>(sum1, sq1, g1, be1, W2, b2, W2p, b2p);
    k_gemm<HID2, HID1><<<BATCH / 64, 256, 0, stream>>>(h1, W2p, b2p, h2, sum2, sq2);
    k_mid2<<<1, 128, 0, stream>>>(sum2, sq2, g2, be2, W3, b3, wb, W3p, b3pp);
    k_final<<<BATCH / 256, 256, 0, stream>>>(h2, W3p, b3pp, wide, (float*)d_out);
}